// GAN_DDI_Block_618475291113
// MI455X (gfx1250) — compile-verified
//
#include <hip/hip_runtime.h>
#include <hip/hip_bf16.h>

// ---------------- problem constants (from reference) ----------------
constexpr int NN  = 100000;        // nodes
constexpr int NE  = 1600000;       // edges
constexpr int NEL = NE + NN;       // edges + self loops
constexpr int FH  = 64;            // F == H*C == 64
constexpr int NH  = 4;             // heads
constexpr int NB  = 512;           // graphs
constexpr float NEG_SLOPE = 0.2f;

typedef float v2f __attribute__((ext_vector_type(2)));
typedef float v8f __attribute__((ext_vector_type(8)));

// ---------------- helpers ----------------
__device__ __forceinline__ float lrelu(float v) {
    return v > 0.f ? v : NEG_SLOPE * v;
}

// float atomic max via signed-max / unsigned-min trick (safe for -inf init, no NaNs)
__device__ __forceinline__ void atomicMaxF(float* addr, float val) {
    if (val >= 0.f)
        atomicMax((int*)addr, __float_as_int(val));
    else
        atomicMin((unsigned int*)addr, (unsigned int)__float_as_int(val));
}

// edge e in the self-loop-augmented list: e < NE -> real edge, else loop
__device__ __forceinline__ void edge_sd(const int* __restrict__ ei, int e, int& s, int& d) {
    if (e < NE) { s = ei[e]; d = ei[NE + e]; }
    else        { s = e - NE; d = e - NE; }
}

// ---------------- K0: init all accumulators ----------------
__global__ void k_init(float* __restrict__ xgat, float* __restrict__ oemb,
                       float* __restrict__ m, float* __restrict__ s,
                       float* __restrict__ score, float* __restrict__ bmax,
                       float* __restrict__ bsum, const float* __restrict__ bias) {
    int i = blockIdx.x * blockDim.x + threadIdx.x;
    if (i < NN * FH) xgat[i] = bias[i & (FH - 1)];   // x_gat accumulates on top of bias
    if (i < NN * NH) { m[i] = -INFINITY; s[i] = 0.f; }
    if (i < NN)      score[i] = 0.f;
    if (i < NB)      { bmax[i] = -INFINITY; bsum[i] = 0.f; }
    if (i < NB * FH) oemb[i] = 0.f;
}

// ---------------- K1: h = x @ W_gat  via f32 WMMA ----------------
// one wave32 per 16-row tile; 4 column tiles of 16 -> full 64 outputs per wave.
// K-loop: 64 / 4 = 16 steps of V_WMMA_F32_16X16X4_F32 per column tile.
__global__ void gemm_h_wmma(const float* __restrict__ x,
                            const float* __restrict__ W,
                            float* __restrict__ h) {
    const int wave = blockIdx.x * (blockDim.x >> 5) + (threadIdx.x >> 5);
    const int lane = threadIdx.x & 31;
    const int row0 = wave * 16;
    if (row0 >= NN) return;              // wave-uniform -> EXEC all ones at WMMA

    const int mr = lane & 15;            // row (A) / col (B,D) within tile
    const int hi = lane >> 4;            // K sub-pair selector (0: K+0..1, 1: K+2..3)

    v8f acc[4] = {};                     // 4 column tiles, 16x16 f32 each

    for (int k = 0; k < FH; k += 4) {
        // A fragment: lane holds x[row0+mr][k + 2*hi], x[row0+mr][k + 2*hi + 1]
        const float* ap = x + (size_t)(row0 + mr) * FH + k + 2 * hi;
        v2f a; a.x = ap[0]; a.y = ap[1];
#pragma unroll
        for (int t = 0; t < 4; ++t) {
            // B fragment: lane holds W[k+2*hi][t*16+mr], W[k+2*hi+1][t*16+mr]
            const float* bp = W + (size_t)(k + 2 * hi) * FH + t * 16 + mr;
            v2f b; b.x = bp[0]; b.y = bp[FH];
            acc[t] = __builtin_amdgcn_wmma_f32_16x16x4_f32(
                /*neg_a=*/false, a, /*neg_b=*/false, b,
                /*c_mod=*/(short)0, acc[t],
                /*reuse_a=*/false, /*reuse_b=*/false);
        }
    }

    // D layout: VGPR j -> row row0 + j + 8*hi, col t*16 + mr
#pragma unroll
    for (int t = 0; t < 4; ++t)
#pragma unroll
        for (int j = 0; j < 8; ++j)
            h[(size_t)(row0 + j + 8 * hi) * FH + t * 16 + mr] = acc[t][j];
}

// ---------------- K2: attention coefficients a_src/a_dst ----------------
__global__ void k_attn(const float* __restrict__ h,
                       const float* __restrict__ attS,
                       const float* __restrict__ attD,
                       float* __restrict__ asrc, float* __restrict__ adst) {
    int i = blockIdx.x * blockDim.x + threadIdx.x;   // n*NH + head
    if (i >= NN * NH) return;
    const int hd = i & (NH - 1);
    const float* hp = h + (size_t)(i >> 2) * FH + hd * 16;
    const float* sp = attS + hd * 16;
    const float* dp = attD + hd * 16;
    float as = 0.f, ad = 0.f;
#pragma unroll
    for (int c = 0; c < 16; ++c) { float v = hp[c]; as += v * sp[c]; ad += v * dp[c]; }
    asrc[i] = as; adst[i] = ad;
}

// ---------------- K3: per-destination per-head max of logits ----------------
__global__ void k_edge_max(const int* __restrict__ ei,
                           const float* __restrict__ asrc, const float* __restrict__ adst,
                           float* __restrict__ m) {
    int e = blockIdx.x * blockDim.x + threadIdx.x;
    if (e >= NEL) return;
    int s, d; edge_sd(ei, e, s, d);
    const float4 as = ((const float4*)asrc)[s];
    const float4 ad = ((const float4*)adst)[d];
    atomicMaxF(&m[d * NH + 0], lrelu(as.x + ad.x));
    atomicMaxF(&m[d * NH + 1], lrelu(as.y + ad.y));
    atomicMaxF(&m[d * NH + 2], lrelu(as.z + ad.z));
    atomicMaxF(&m[d * NH + 3], lrelu(as.w + ad.w));
}

// ---------------- K4: per-destination per-head sum of exp ----------------
__global__ void k_edge_expsum(const int* __restrict__ ei,
                              const float* __restrict__ asrc, const float* __restrict__ adst,
                              const float* __restrict__ m, float* __restrict__ sum) {
    int e = blockIdx.x * blockDim.x + threadIdx.x;
    if (e >= NEL) return;
    int s, d; edge_sd(ei, e, s, d);
    const float4 as = ((const float4*)asrc)[s];
    const float4 ad = ((const float4*)adst)[d];
    const float4 mm = ((const float4*)m)[d];
    atomicAdd(&sum[d * NH + 0], expf(lrelu(as.x + ad.x) - mm.x));
    atomicAdd(&sum[d * NH + 1], expf(lrelu(as.y + ad.y) - mm.y));
    atomicAdd(&sum[d * NH + 2], expf(lrelu(as.z + ad.z) - mm.z));
    atomicAdd(&sum[d * NH + 3], expf(lrelu(as.w + ad.w) - mm.w));
}

// ---------------- K5: weighted message scatter into x_gat ----------------
__global__ void k_edge_msg(const int* __restrict__ ei,
                           const float* __restrict__ asrc, const float* __restrict__ adst,
                           const float* __restrict__ m, const float* __restrict__ sum,
                           const float* __restrict__ h, float* __restrict__ xgat) {
    int e = blockIdx.x * blockDim.x + threadIdx.x;
    if (e >= NEL) return;
    int s, d; edge_sd(ei, e, s, d);
    const float4 as = ((const float4*)asrc)[s];
    const float4 ad = ((const float4*)adst)[d];
    const float4 mm = ((const float4*)m)[d];
    const float4 ss = ((const float4*)sum)[d];
    float alpha[NH];
    alpha[0] = expf(lrelu(as.x + ad.x) - mm.x) / ss.x;
    alpha[1] = expf(lrelu(as.y + ad.y) - mm.y) / ss.y;
    alpha[2] = expf(lrelu(as.z + ad.z) - mm.z) / ss.z;
    alpha[3] = expf(lrelu(as.w + ad.w) - mm.w) / ss.w;

    const float4* hp = (const float4*)(h + (size_t)s * FH);
    float* op = xgat + (size_t)d * FH;
#pragma unroll
    for (int q = 0; q < 16; ++q) {               // q*4 channels; head = q/4
        float4 v = hp[q];
        float a = alpha[q >> 2];
        atomicAdd(&op[q * 4 + 0], v.x * a);
        atomicAdd(&op[q * 4 + 1], v.y * a);
        atomicAdd(&op[q * 4 + 2], v.z * a);
        atomicAdd(&op[q * 4 + 3], v.w * a);
    }
}

// ---------------- K6: GraphConv relational score (original edges only) ----------------
__global__ void k_edge_rel(const int* __restrict__ ei, const float* __restrict__ xgat,
                           const float* __restrict__ wrel, float* __restrict__ score) {
    int e = blockIdx.x * blockDim.x + threadIdx.x;
    if (e >= NE) return;
    const int s = ei[e], d = ei[NE + e];
    const float4* xp = (const float4*)(xgat + (size_t)s * FH);
    const float4* wp = (const float4*)wrel;
    float acc = 0.f;
#pragma unroll
    for (int q = 0; q < 16; ++q) {
        float4 v = xp[q]; float4 w = wp[q];
        acc += v.x * w.x + v.y * w.y + v.z * w.z + v.w * w.w;
    }
    atomicAdd(&score[d], acc);
}

// ---------------- K7: add root term + bias to score ----------------
__global__ void k_node_score(const float* __restrict__ xgat, const float* __restrict__ wroot,
                             const float* __restrict__ bsc, float* __restrict__ score) {
    int n = blockIdx.x * blockDim.x + threadIdx.x;
    if (n >= NN) return;
    const float4* xp = (const float4*)(xgat + (size_t)n * FH);
    const float4* wp = (const float4*)wroot;
    float acc = bsc[0];
#pragma unroll
    for (int q = 0; q < 16; ++q) {
        float4 v = xp[q]; float4 w = wp[q];
        acc += v.x * w.x + v.y * w.y + v.z * w.z + v.w * w.w;
    }
    score[n] += acc;
}

// ---------------- K8/K9: softmax over batch segments ----------------
__global__ void k_batch_max(const int* __restrict__ batch, const float* __restrict__ score,
                            float* __restrict__ bmax) {
    int n = blockIdx.x * blockDim.x + threadIdx.x;
    if (n >= NN) return;
    atomicMaxF(&bmax[batch[n]], score[n]);
}

__global__ void k_batch_exp(const int* __restrict__ batch, float* __restrict__ score,
                            const float* __restrict__ bmax, float* __restrict__ bsum) {
    int n = blockIdx.x * blockDim.x + threadIdx.x;
    if (n >= NN) return;
    const int b = batch[n];
    float e = expf(score[n] - bmax[b]);
    score[n] = e;                        // stash numerator
    atomicAdd(&bsum[b], e);
}

// ---------------- K10: scale by softmax score + global_add_pool ----------------
__global__ void k_pool(const int* __restrict__ batch, const float* __restrict__ score,
                       const float* __restrict__ bsum, const float* __restrict__ xgat,
                       float* __restrict__ oemb) {
    int n = blockIdx.x * blockDim.x + threadIdx.x;
    if (n >= NN) return;
    const int b = batch[n];
    const float sc = score[n] / bsum[b];
    const float4* xp = (const float4*)(xgat + (size_t)n * FH);
    float* op = oemb + (size_t)b * FH;
#pragma unroll
    for (int q = 0; q < 16; ++q) {
        float4 v = xp[q];
        atomicAdd(&op[q * 4 + 0], v.x * sc);
        atomicAdd(&op[q * 4 + 1], v.y * sc);
        atomicAdd(&op[q * 4 + 2], v.z * sc);
        atomicAdd(&op[q * 4 + 3], v.w * sc);
    }
}

// ---------------- launch ----------------
extern "C" void kernel_launch(void* const* d_in, const int* in_sizes, int n_in,
                              void* d_out, int out_size, void* d_ws, size_t ws_size,
                              hipStream_t stream) {
    (void)in_sizes; (void)n_in; (void)out_size; (void)ws_size;

    const float* x     = (const float*)d_in[0];
    const int*   ei    = (const int*)  d_in[1];   // [2,E] flat: src @ [0,E), dst @ [E,2E)
    const int*   batch = (const int*)  d_in[2];
    const float* Wgat  = (const float*)d_in[3];
    const float* attS  = (const float*)d_in[4];
    const float* attD  = (const float*)d_in[5];
    const float* bias  = (const float*)d_in[6];
    const float* wrel  = (const float*)d_in[7];
    const float* wroot = (const float*)d_in[8];
    const float* bsc   = (const float*)d_in[9];

    float* xgat = (float*)d_out;                 // [N,64]
    float* oemb = xgat + (size_t)NN * FH;        // [B,64]

    float* ws   = (float*)d_ws;                  // ~32.5 MB total
    float* h    = ws;                            // [N,64]
    float* asrc = h    + (size_t)NN * FH;        // [N,4]
    float* adst = asrc + (size_t)NN * NH;        // [N,4]
    float* mbuf = adst + (size_t)NN * NH;        // [N,4]
    float* sbuf = mbuf + (size_t)NN * NH;        // [N,4]
    float* scr  = sbuf + (size_t)NN * NH;        // [N]
    float* bmax = scr  + (size_t)NN;             // [B]
    float* bsum = bmax + (size_t)NB;             // [B]

    const int T = 256;
    k_init<<<(NN * FH + T - 1) / T, T, 0, stream>>>(xgat, oemb, mbuf, sbuf, scr, bmax, bsum, bias);

    // 6250 row-tiles, 4 waves per 128-thread block
    const int tiles = NN / 16;
    gemm_h_wmma<<<(tiles + 3) / 4, 128, 0, stream>>>(x, Wgat, h);

    k_attn      <<<(NN * NH + T - 1) / T, T, 0, stream>>>(h, attS, attD, asrc, adst);
    k_edge_max  <<<(NEL + T - 1) / T, T, 0, stream>>>(ei, asrc, adst, mbuf);
    k_edge_expsum<<<(NEL + T - 1) / T, T, 0, stream>>>(ei, asrc, adst, mbuf, sbuf);
    k_edge_msg  <<<(NEL + T - 1) / T, T, 0, stream>>>(ei, asrc, adst, mbuf, sbuf, h, xgat);
    k_edge_rel  <<<(NE + T - 1) / T, T, 0, stream>>>(ei, xgat, wrel, scr);
    k_node_score<<<(NN + T - 1) / T, T, 0, stream>>>(xgat, wroot, bsc, scr);
    k_batch_max <<<(NN + T - 1) / T, T, 0, stream>>>(batch, scr, bmax);
    k_batch_exp <<<(NN + T - 1) / T, T, 0, stream>>>(batch, scr, bmax, bsum);
    k_pool      <<<(NN + T - 1) / T, T, 0, stream>>>(batch, scr, bsum, xgat, oemb);
}